// VectorQuantizer_4853313044735
// MI455X (gfx1250) — compile-verified
//
#include <hip/hip_runtime.h>
#include <math.h>

// ---------------------------------------------------------------------------
// VectorQuantizer for MI455X (gfx1250, wave32).
//
// Score matrix xn @ en^T: 2*131072*2048*64 = 3.4e10 FLOPs over ~65MB of HBM
// traffic (AI ~520 flop/byte) -> compute bound -> matrix pipe required.
// fp32 WMMA (v_wmma_f32_16x16x4_f32) keeps full fp32 precision for the argmin
// (nearest-neighbor flips from low-precision rounding would corrupt idx).
//
// - Each wave owns TWO 16-row A tiles (64 VGPRs, LDS-resident x tile) sharing
//   every B fragment -> 32 WMMAs per chunk, two independent D->C chains.
// - Codebook is pre-packed into WMMA-B fragment order so a 16-codeword chunk
//   is fetched with 8 coalesced global_load_b128 into a live v4f[8] array
//   (forces the allocator to keep all 8 loads in flight: one clause + one
//   wait per 32 WMMAs, instead of load/wait/wmma serialization).
// ---------------------------------------------------------------------------

typedef __attribute__((ext_vector_type(2))) float v2f;
typedef __attribute__((ext_vector_type(4))) float v4f;
typedef __attribute__((ext_vector_type(8))) float v8f;

#define N_ROWS   131072
#define DIM      64
#define KCODE    2048
#define TILE_M   256          // x-rows per block (8 waves * 32 rows)
#define LDS_STR  68           // padded row stride (floats): 68 mod 64 = 4 banks
#define NBLOCKS  (N_ROWS / TILE_M)   // 512
#define NCHUNK   (KCODE / 16)        // 128 codeword chunks

// ---------------------------------------------------------------------------
// Kernel A: normalize codebook rows; en[k] = e[k]/max(||e||,eps); en2[k]=sum(en^2)
// One wave (32 lanes) per codebook row; lane l covers elements l and l+32.
// ---------------------------------------------------------------------------
__global__ __launch_bounds__(32)
void vq_norm_embed(const float* __restrict__ e,
                   float* __restrict__ en,
                   float* __restrict__ en2) {
    const int k = blockIdx.x;
    const int l = threadIdx.x;
    float a = e[(size_t)k * DIM + l];
    float b = e[(size_t)k * DIM + 32 + l];
    float s = a * a + b * b;
    #pragma unroll
    for (int m = 16; m >= 1; m >>= 1) s += __shfl_xor(s, m, 32);
    const float inv = 1.0f / fmaxf(sqrtf(s), 1e-12f);
    const float na = a * inv, nb = b * inv;
    en[(size_t)k * DIM + l]      = na;
    en[(size_t)k * DIM + 32 + l] = nb;
    float s2 = na * na + nb * nb;
    #pragma unroll
    for (int m = 16; m >= 1; m >>= 1) s2 += __shfl_xor(s2, m, 32);
    if (l == 0) en2[k] = s2;
}

// ---------------------------------------------------------------------------
// Kernel A2: repack normalized codebook into WMMA-B fragment order.
// For chunk c (16 codewords), lane l of a wave needs, per K-step kk:
//   en[c*16 + (l&15)][kk*4 + (l>>4)*2 .. +1]
// We store those 16 float2 contiguously (128B per lane, 4KB per chunk) so the
// main loop fetches a chunk's B matrix with 8 coalesced global_load_b128.
// ---------------------------------------------------------------------------
__global__ __launch_bounds__(32)
void vq_pack_b(const float* __restrict__ en, float* __restrict__ enp) {
    const int c = blockIdx.x;           // 128 chunks
    const int l = threadIdx.x;
    const int n     = c * 16 + (l & 15);
    const int khalf = (l >> 4) * 2;
    float* dst = enp + (size_t)c * (32 * 32) + (size_t)l * 32;  // 128B/lane
    #pragma unroll
    for (int kk = 0; kk < 16; ++kk) {
        v2f b = *(const v2f*)&en[(size_t)n * DIM + kk * 4 + khalf];
        *(v2f*)&dst[kk * 2] = b;
    }
}

// ---------------------------------------------------------------------------
// Kernel B: per block: normalize 256 x-rows in LDS, WMMA-sweep the codebook
// (2 M-tiles per wave), argmin per row, emit quantized rows + idx + partial.
// ---------------------------------------------------------------------------
__global__ __launch_bounds__(256)
void vq_main(const float* __restrict__ x,
             const float* __restrict__ en,
             const float* __restrict__ enp,
             const float* __restrict__ en2,
             float* __restrict__ out_q,      // [N, 64]
             float* __restrict__ out_idx,    // [N] (stored as float)
             float* __restrict__ partials) { // [NBLOCKS]
    __shared__ float lxn[TILE_M * LDS_STR];
    __shared__ int   lidx[TILE_M];
    __shared__ float lred[256];

    const int tid     = threadIdx.x;
    const int lane    = tid & 31;
    const int wave    = tid >> 5;
    const int rowbase = blockIdx.x * TILE_M;

    // ---- 1) coalesced float4 load of the x tile into padded LDS ----
    for (int i = tid; i < TILE_M * (DIM / 4); i += 256) {
        const int r  = i >> 4;        // i / 16
        const int e4 = i & 15;
        v4f v = ((const v4f*)x)[(size_t)(rowbase + r) * (DIM / 4) + e4];
        *(v4f*)&lxn[r * LDS_STR + e4 * 4] = v;   // 272B row stride: 16B aligned
    }
    __syncthreads();

    // ---- 2) L2-normalize each row in LDS (one thread per row) ----
    {
        float s = 0.f;
        #pragma unroll
        for (int e = 0; e < DIM; ++e) { float t = lxn[tid * LDS_STR + e]; s += t * t; }
        const float inv = 1.0f / fmaxf(sqrtf(s), 1e-12f);
        #pragma unroll
        for (int e = 0; e < DIM; ++e) lxn[tid * LDS_STR + e] *= inv;
    }
    __syncthreads();

    // ---- 3) A fragments for two 16-row tiles (fp32 WMMA 16x4 layout) ----
    // lanes 0-15: M=lane, V0=K0,V1=K1 ; lanes 16-31: M=lane-16, V0=K2,V1=K3
    const int wrow  = wave * 32;              // this wave's 32-row slab
    const int m16   = lane & 15;
    const int khalf = (lane >> 4) * 2;        // 0 or 2
    v2f afrag0[16], afrag1[16];
    #pragma unroll
    for (int kk = 0; kk < 16; ++kk) {
        afrag0[kk] = *(const v2f*)&lxn[(wrow +      m16) * LDS_STR + kk * 4 + khalf];
        afrag1[kk] = *(const v2f*)&lxn[(wrow + 16 + m16) * LDS_STR + kk * 4 + khalf];
    }

    // ---- 4) sweep codebook: 8 x b128 loads then 32 WMMAs per chunk ----
    float bestv0[8], bestv1[8];
    int   besti0[8], besti1[8];
    #pragma unroll
    for (int v = 0; v < 8; ++v) {
        bestv0[v] = __builtin_inff(); besti0[v] = 0;
        bestv1[v] = __builtin_inff(); besti1[v] = 0;
    }

    const v4f* bp = (const v4f*)enp;
    for (int c = 0; c < NCHUNK; ++c) {
        const int nb = c * 16;
        // whole-chunk B fragment fetch: 8 live b128 loads -> single clause
        v4f bq[8];
        const size_t bbase = (size_t)c * 256 + (size_t)lane * 8;  // v4f units
        #pragma unroll
        for (int j = 0; j < 8; ++j) bq[j] = bp[bbase + j];
        const float e2 = en2[nb + m16];

        v8f acc0 = {};
        v8f acc1 = {};
        #pragma unroll
        for (int kk = 0; kk < 16; ++kk) {
            const v4f bw = bq[kk >> 1];
            const v2f b  = (kk & 1) ? __builtin_shufflevector(bw, bw, 2, 3)
                                    : __builtin_shufflevector(bw, bw, 0, 1);
            acc0 = __builtin_amdgcn_wmma_f32_16x16x4_f32(
                     false, afrag0[kk], false, b, (short)0, acc0, false, false);
            acc1 = __builtin_amdgcn_wmma_f32_16x16x4_f32(
                     false, afrag1[kk], false, b, (short)0, acc1, false, false);
        }
        // C layout: VGPR v, lane L -> row (v + 8*(L/16)), col N = L%16.
        // dist = ||en||^2 - 2*dot  (+||xn||^2 is per-row constant: argmin-invariant)
        #pragma unroll
        for (int v = 0; v < 8; ++v) {
            const float d0 = e2 - 2.0f * acc0[v];
            if (d0 < bestv0[v]) { bestv0[v] = d0; besti0[v] = nb + m16; }
            const float d1 = e2 - 2.0f * acc1[v];
            if (d1 < bestv1[v]) { bestv1[v] = d1; besti1[v] = nb + m16; }
        }
    }

    // ---- 5) butterfly min across the 16 N-lanes (tie -> smaller idx) ----
    #pragma unroll
    for (int t = 0; t < 2; ++t) {
        #pragma unroll
        for (int v = 0; v < 8; ++v) {
            float bv = t ? bestv1[v] : bestv0[v];
            int   bi = t ? besti1[v] : besti0[v];
            #pragma unroll
            for (int m = 1; m <= 8; m <<= 1) {
                const float ov = __shfl_xor(bv, m, 32);
                const int   oi = __shfl_xor(bi, m, 32);
                if (ov < bv || (ov == bv && oi < bi)) { bv = ov; bi = oi; }
            }
            if (m16 == 0) {        // lane 0 -> row +v ; lane 16 -> row +v+8
                const int r = wrow + t * 16 + v + ((lane >> 4) << 3);
                lidx[r] = bi;
                out_idx[rowbase + r] = (float)bi;
            }
        }
    }
    __syncthreads();

    // ---- 6) gather quantized rows, write out, accumulate loss partial ----
    float part = 0.f;
    for (int i = tid; i < TILE_M * (DIM / 4); i += 256) {
        const int r  = i >> 4;
        const int e4 = i & 15;
        const int bi = lidx[r];
        v4f q  = ((const v4f*)en)[(size_t)bi * (DIM / 4) + e4];
        v4f xv = *(const v4f*)&lxn[r * LDS_STR + e4 * 4];
        v4f d  = q - xv;
        part += d.x * d.x + d.y * d.y + d.z * d.z + d.w * d.w;
        ((v4f*)out_q)[(size_t)(rowbase + r) * (DIM / 4) + e4] = q;
    }
    lred[tid] = part;
    __syncthreads();
    #pragma unroll
    for (int s = 128; s > 0; s >>= 1) {
        if (tid < s) lred[tid] += lred[tid + s];
        __syncthreads();
    }
    if (tid == 0) partials[blockIdx.x] = lred[0];
}

// ---------------------------------------------------------------------------
// Kernel C: deterministic fixed-order reduction of block partials -> loss.
// loss = q_latent + 0.25*e_latent = 1.25 * mean((q - xn)^2). Plain store
// (not atomic) overwrites the harness's poison in the loss slot, and the
// fixed tree order keeps the result bitwise-identical across graph replays.
// ---------------------------------------------------------------------------
__global__ __launch_bounds__(256)
void vq_loss_reduce(const float* __restrict__ partials, float* __restrict__ loss) {
    __shared__ float lred[256];
    const int t = threadIdx.x;
    float s = 0.f;
    for (int i = t; i < NBLOCKS; i += 256) s += partials[i];
    lred[t] = s;
    __syncthreads();
    #pragma unroll
    for (int k = 128; k > 0; k >>= 1) {
        if (t < k) lred[t] += lred[t + k];
        __syncthreads();
    }
    if (t == 0) *loss = lred[0] * 1.25f / ((float)N_ROWS * (float)DIM);
}

extern "C" void kernel_launch(void* const* d_in, const int* in_sizes, int n_in,
                              void* d_out, int out_size, void* d_ws, size_t ws_size,
                              hipStream_t stream) {
    const float* x = (const float*)d_in[0];          // [131072, 64]
    const float* e = (const float*)d_in[1];          // [2048, 64]

    float* out      = (float*)d_out;
    float* out_q    = out;                            // [N*64]
    float* out_loss = out + (size_t)N_ROWS * DIM;     // [1]
    float* out_idx  = out_loss + 1;                   // [N]

    float* wsf      = (float*)d_ws;
    float* en       = wsf;                            // [2048*64]
    float* en2      = en + (size_t)KCODE * DIM;       // [2048]
    float* partials = en2 + KCODE;                    // [NBLOCKS]
    float* enp      = partials + NBLOCKS;             // [2048*64] packed B frags

    vq_norm_embed<<<KCODE, 32, 0, stream>>>(e, en, en2);
    vq_pack_b<<<NCHUNK, 32, 0, stream>>>(en, enp);
    vq_main<<<NBLOCKS, 256, 0, stream>>>(x, en, enp, en2, out_q, out_idx, partials);
    vq_loss_reduce<<<1, 256, 0, stream>>>(partials, out_loss);
}